// MomentsSpoke_52450140619338
// MI455X (gfx1250) — compile-verified
//
#include <hip/hip_runtime.h>
#include <math.h>

// MomentsSpoke: per-segment (8 x 512) moments [mean, std(ddof=1), skew, kurt]
// -> [B,32] features -> Linear(32,32) via V_WMMA_F32_16X16X4_F32.
//
// HBM-bound: 1.07 GB read @ 23.3 TB/s (~46us floor). This version maximizes
// memory parallelism: 4096 blocks (one per 16-row tile) x 8 waves = 32768
// streaming waves, each wave owning 16 segments with paired (double-buffered)
// non-temporal b128 loads. Wave 0 of each block runs the 16-WMMA linear tail.

#define D_DIM        4096
#define SEGLEN       512
#define ROWS_PER_BLK 16
#define SMEM_STRIDE  34   // even pad: float2 LDS reads stay 8B aligned,
                          // M*34 mod 64 distinct for M=0..15 -> conflict-free

typedef __attribute__((ext_vector_type(2))) float v2f;
typedef __attribute__((ext_vector_type(4))) float v4f;
typedef __attribute__((ext_vector_type(8))) float v8f;

__device__ __forceinline__ void seg_moments(const v4f v[4], int lane,
                                            float* __restrict__ dst) {
  // pass 1: mean (512 elems live in 16 regs/lane)
  float s1 = 0.f;
#pragma unroll
  for (int t = 0; t < 4; ++t) s1 += (v[t][0] + v[t][1]) + (v[t][2] + v[t][3]);
#pragma unroll
  for (int off = 16; off > 0; off >>= 1) s1 += __shfl_xor(s1, off, 32);
  const float mean = s1 * (1.0f / 512.0f);

  // pass 2: centered sums, matching the reference's two-pass math
  float s2 = 0.f, s3 = 0.f, s4 = 0.f;
#pragma unroll
  for (int t = 0; t < 4; ++t) {
#pragma unroll
    for (int j = 0; j < 4; ++j) {
      float c = v[t][j] - mean, c2 = c * c;
      s2 += c2; s3 += c2 * c; s4 += c2 * c2;
    }
  }
#pragma unroll
  for (int off = 16; off > 0; off >>= 1) {
    s2 += __shfl_xor(s2, off, 32);
    s3 += __shfl_xor(s3, off, 32);
    s4 += __shfl_xor(s4, off, 32);
  }

  if (lane == 0) {
    const float var = s2 * (1.0f / 511.0f);             // ddof = 1
    const float sd  = sqrtf(var) + 1e-8f;
    const float sd2 = sd * sd;
    dst[0] = mean;
    dst[1] = sd;
    dst[2] = (s3 * (1.0f / 512.0f)) / (sd2 * sd);
    dst[3] = (s4 * (1.0f / 512.0f)) / (sd2 * sd2) - 3.0f;
  }
}

__global__ __launch_bounds__(256)
void moments_linear_kernel(const float* __restrict__ x,
                           const float* __restrict__ W,     // [32 out][32 in]
                           const float* __restrict__ bias,  // [32]
                           float* __restrict__ out,         // [rows][32]
                           int rows) {
  __shared__ float smem[ROWS_PER_BLK * SMEM_STRIDE];

  const int lane = threadIdx.x & 31;
  const int wave = threadIdx.x >> 5;
  const int r0   = blockIdx.x * ROWS_PER_BLK;

  // ------------- Phase 1: 8 waves x 16 segments (2 rows per wave) -----------
  const int q0 = wave * 16;                 // segment ids q0..q0+15 (q = rloc*8+seg)
  for (int s = 0; s < 16; s += 2) {         // paired: 8 x b128 in flight per lane
    const int qa = q0 + s, qb = q0 + s + 1;
    const int ra = r0 + (qa >> 3), rb = r0 + (qb >> 3);
    const bool oka = ra < rows, okb = rb < rows;   // wave-uniform

    v4f va[4], vb[4];
    const float* pa = x + (size_t)ra * D_DIM + (qa & 7) * SEGLEN;
    const float* pb = x + (size_t)rb * D_DIM + (qb & 7) * SEGLEN;
    if (oka) {
#pragma unroll
      for (int t = 0; t < 4; ++t)
        va[t] = __builtin_nontemporal_load((const v4f*)(pa + (size_t)(t * 32 + lane) * 4));
    }
    if (okb) {
#pragma unroll
      for (int t = 0; t < 4; ++t)
        vb[t] = __builtin_nontemporal_load((const v4f*)(pb + (size_t)(t * 32 + lane) * 4));
    }
    if (oka) seg_moments(va, lane, &smem[(qa >> 3) * SMEM_STRIDE + (qa & 7) * 4]);
    if (okb) seg_moments(vb, lane, &smem[(qb >> 3) * SMEM_STRIDE + (qb & 7) * 4]);
  }

  __syncthreads();

  // ------------- Phase 2 (wave 0): 16x32 tile = moments @ W^T + b -----------
  if (threadIdx.x < 32) {                   // execz-skip for waves 1..7;
    const int l16  = lane & 15;             // executing wave has EXEC all-1s
    const int half = lane >> 4;
    const int kb   = half * 2;              // f32 16x16x4 K sub-lane layout

    v8f acc0, acc1;
    const float b0 = bias[l16], b1 = bias[16 + l16];
#pragma unroll
    for (int r = 0; r < 8; ++r) { acc0[r] = b0; acc1[r] = b1; }

#pragma unroll
    for (int kk = 0; kk < 8; ++kk) {
      const int k = kk * 4 + kb;
      v2f a, bv0, bv1;
      a[0]   = smem[l16 * SMEM_STRIDE + k];  a[1]   = smem[l16 * SMEM_STRIDE + k + 1];
      bv0[0] = W[l16 * 32 + k];              bv0[1] = W[l16 * 32 + k + 1];  // B[k][N]=W[N][k]
      bv1[0] = W[(16 + l16) * 32 + k];       bv1[1] = W[(16 + l16) * 32 + k + 1];
      acc0 = __builtin_amdgcn_wmma_f32_16x16x4_f32(false, a, false, bv0, (short)0, acc0, false, false);
      acc1 = __builtin_amdgcn_wmma_f32_16x16x4_f32(false, a, false, bv1, (short)0, acc1, false, false);
    }

#pragma unroll
    for (int r = 0; r < 8; ++r) {           // D: VGPR r -> M = r + 8*half, N = l16
      const int row = r0 + r + half * 8;
      if (row < rows) {
        out[(size_t)row * 32 + l16]      = acc0[r];
        out[(size_t)row * 32 + 16 + l16] = acc1[r];
      }
    }
  }
}

extern "C" void kernel_launch(void* const* d_in, const int* in_sizes, int n_in,
                              void* d_out, int out_size, void* d_ws, size_t ws_size,
                              hipStream_t stream) {
  (void)n_in; (void)out_size; (void)d_ws; (void)ws_size;
  const float* x    = (const float*)d_in[0];
  const float* W    = (const float*)d_in[1];
  const float* bias = (const float*)d_in[2];
  float* out = (float*)d_out;

  const int rows   = in_sizes[0] / D_DIM;                       // 65536
  const int blocks = (rows + ROWS_PER_BLK - 1) / ROWS_PER_BLK;  // 4096
  moments_linear_kernel<<<blocks, 256, 0, stream>>>(x, W, bias, out, rows);
}